// AttentionPool_29119878267067
// MI455X (gfx1250) — compile-verified
//
#include <hip/hip_runtime.h>
#include <hip/hip_bf16.h>
#include <math.h>

// Problem constants (from reference)
#define NB   32          // N batch
#define CH   256         // C channels
#define TT   512         // T frames
#define VV   25          // V joints
#define HH   8           // heads
#define HD   32          // head dim (C/H)
#define C2   512         // 2C
#define NTP  128         // new_t = ceil(T/4)
#define S0   0.02209708691207961f   // 1/(8*sqrt(32))

typedef float v2f __attribute__((ext_vector_type(2)));
typedef float v8f __attribute__((ext_vector_type(8)));

// ---------------------------------------------------------------------------
// K1: xn[(n*T+t)*C + c] = mean_v x[n,c,t,v]   (streams all of x: ~419 MB)
// ---------------------------------------------------------------------------
__global__ void k_mean_v(const float* __restrict__ x, float* __restrict__ xn) {
    const unsigned total = NB * CH * TT;                 // 4,194,304
    unsigned idx = blockIdx.x * blockDim.x + threadIdx.x;
    if (idx >= total) return;
    const float* p = x + (size_t)idx * VV;
    __builtin_prefetch(p + 64 * VV, 0, 1);               // global_prefetch_b8
    float s = 0.f;
#pragma unroll
    for (int v = 0; v < VV; ++v) s += p[v];
    unsigned t = idx % TT;
    unsigned c = (idx / TT) % CH;
    unsigned n = idx / (TT * CH);
    xn[((size_t)n * TT + t) * CH + c] = s * (1.0f / VV);
}

// ---------------------------------------------------------------------------
// K2: srow[n][c] = sum_t xn[n,t,c]
// ---------------------------------------------------------------------------
__global__ void k_colsum(const float* __restrict__ xn, float* __restrict__ srow) {
    int n = blockIdx.x;          // 32
    int c = threadIdx.x;         // 256
    float s = 0.f;
    for (int t = 0; t < TT; ++t) s += xn[((size_t)n * TT + t) * CH + c];
    srow[n * CH + c] = s;
}

// ---------------------------------------------------------------------------
// K3: kbar[n][j] = (1/T) * (srow[n,:] . W[:, colk(j)]) + b[colk(j)]
//     colk(j) = (j/32)*64 + 32 + (j%32)
// ---------------------------------------------------------------------------
__global__ void k_kbar(const float* __restrict__ srow, const float* __restrict__ W,
                       const float* __restrict__ b, float* __restrict__ kbar) {
    int n = blockIdx.x;          // 32
    int j = threadIdx.x;         // 256
    int col = ((j >> 5) << 6) + 32 + (j & 31);
    float s = 0.f;
    for (int c = 0; c < CH; ++c) s += srow[n * CH + c] * W[(size_t)c * C2 + col];
    kbar[n * CH + j] = s * (1.0f / TT) + b[col];
}

// ---------------------------------------------------------------------------
// K4: q = xn @ Wq + bq via V_WMMA_F32_16X16X4_F32.
//     One wave computes a 16x16 tile of q (rows = n*T+t, cols = j=0..255),
//     looping K=256 in steps of 4.
//     A 16x4 layout : lane<16 -> M=lane, K=k..k+1 ; lane>=16 -> K=k+2..k+3
//     B 4x16 layout : lane<16 -> N=lane, K=k..k+1 ; lane>=16 -> K=k+2..k+3
//     C/D layout    : vgpr r -> M=r (lanes 0-15) / M=r+8 (lanes 16-31), N=lane%16
// ---------------------------------------------------------------------------
__global__ void k_qproj_wmma(const float* __restrict__ xn, const float* __restrict__ W,
                             const float* __restrict__ b, float* __restrict__ q) {
    unsigned wave = (blockIdx.x * blockDim.x + threadIdx.x) >> 5;   // 16384 waves
    unsigned lane = threadIdx.x & 31;
    unsigned colTile = wave & 15;          // 16 col tiles (256 cols / 16)
    unsigned rowTile = wave >> 4;          // 1024 row tiles (16384 rows / 16)
    unsigned row0 = rowTile * 16;
    unsigned half = lane >> 4;             // 0: K lo-pair, 1: K hi-pair
    unsigned m    = lane & 15;             // M for A, N for B/D
    unsigned jl   = colTile * 16 + m;                       // logical q column
    unsigned col  = ((jl >> 5) << 6) + (jl & 31);           // colq(j) in W/b

    float bias = b[col];
    v8f acc;
#pragma unroll
    for (int i = 0; i < 8; ++i) acc[i] = bias;              // fold bias into C

    const float* arow = xn + (size_t)(row0 + m) * CH;       // A row for this lane
    const float* bcol = W + col;                            // B column
#pragma unroll 4
    for (int k = 0; k < CH; k += 4) {
        int ka = k + (int)(half << 1);
        v2f a = *(const v2f*)(arow + ka);                   // A[m][ka], A[m][ka+1]
        v2f bb;
        bb.x = bcol[(size_t)ka * C2];                       // B[ka][m]
        bb.y = bcol[(size_t)(ka + 1) * C2];                 // B[ka+1][m]
        acc = __builtin_amdgcn_wmma_f32_16x16x4_f32(
            /*neg_a=*/false, a, /*neg_b=*/false, bb,
            /*c_mod=*/(short)0, acc, /*reuse_a=*/false, /*reuse_b=*/false);
    }

#pragma unroll
    for (int r = 0; r < 8; ++r) {
        unsigned M = r + half * 8;
        q[(size_t)(row0 + M) * CH + jl] = acc[r];
    }
}

// ---------------------------------------------------------------------------
// K5: scores[row] = S0 * (q[row,:] . kbar[n,:]),  row = n*T+t  (wave per row)
// ---------------------------------------------------------------------------
__global__ void k_scores(const float* __restrict__ q, const float* __restrict__ kbar,
                         float* __restrict__ scores) {
    unsigned row  = (blockIdx.x * blockDim.x + threadIdx.x) >> 5;   // 16384
    unsigned lane = threadIdx.x & 31;
    unsigned n = row >> 9;                                          // row / T
    const float* qr = q + (size_t)row * CH;
    const float* kb = kbar + n * CH;
    float s = 0.f;
#pragma unroll
    for (int j = lane; j < CH; j += 32) s += qr[j] * kb[j];
#pragma unroll
    for (int off = 16; off > 0; off >>= 1) s += __shfl_xor(s, off, 32);
    if (lane == 0) scores[row] = s * S0;
}

// ---------------------------------------------------------------------------
// K6: exact top-k by rank counting (matches lax.top_k: descending, stable ties)
// ---------------------------------------------------------------------------
__global__ void k_topk(const float* __restrict__ scores,
                       float* __restrict__ topval, int* __restrict__ topidx) {
    __shared__ float s[TT];
    int n = blockIdx.x;          // 32
    int t = threadIdx.x;         // 512
    float v = scores[n * TT + t];
    s[t] = v;
    __syncthreads();
    int rank = 0;
    for (int u = 0; u < TT; ++u) {
        float su = s[u];
        rank += (su > v) || (su == v && u < t);
    }
    if (rank < NTP) {
        topval[n * NTP + rank] = v;
        topidx[n * NTP + rank] = t;
    }
}

// ---------------------------------------------------------------------------
// K7: out[n,c,r,v] = x[n,c,idx[n][r],v] * sigmoid(topval[n][r])
// ---------------------------------------------------------------------------
__global__ void k_gather(const float* __restrict__ x, const float* __restrict__ topval,
                         const int* __restrict__ topidx, float* __restrict__ out) {
    const unsigned total = NB * CH * NTP * VV;              // 26,214,400
    unsigned idx = blockIdx.x * blockDim.x + threadIdx.x;
    if (idx >= total) return;
    unsigned v = idx % VV;
    unsigned rest = idx / VV;
    unsigned r = rest % NTP;  rest /= NTP;
    unsigned c = rest % CH;
    unsigned n = rest / CH;
    int t = topidx[n * NTP + r];
    float g = 1.0f / (1.0f + __expf(-topval[n * NTP + r]));
    out[idx] = x[(((size_t)n * CH + c) * TT + t) * VV + v] * g;
}

// ---------------------------------------------------------------------------
extern "C" void kernel_launch(void* const* d_in, const int* in_sizes, int n_in,
                              void* d_out, int out_size, void* d_ws, size_t ws_size,
                              hipStream_t stream) {
    const float* x = (const float*)d_in[0];   // (N, C, T, V)
    const float* W = (const float*)d_in[1];   // (C, 2C)
    const float* b = (const float*)d_in[2];   // (2C,)
    float* out = (float*)d_out;               // (N, C, new_t, V)

    // workspace layout (floats)
    float* ws = (float*)d_ws;
    const size_t NROWS = (size_t)NB * TT;           // 16384
    float* xn     = ws;                             // NROWS*CH = 4,194,304
    float* q      = xn + NROWS * CH;                // 4,194,304
    float* srow   = q + NROWS * CH;                 // 8192
    float* kbar   = srow + NB * CH;                 // 8192
    float* scores = kbar + NB * CH;                 // 16384
    float* topval = scores + NROWS;                 // 4096
    int*   topidx = (int*)(topval + NB * NTP);      // 4096

    // 1) V-mean (streams x)
    {
        unsigned total = NB * CH * TT;
        k_mean_v<<<(total + 255) / 256, 256, 0, stream>>>(x, xn);
    }
    // 2) time-sum per channel
    k_colsum<<<NB, CH, 0, stream>>>(xn, srow);
    // 3) kbar
    k_kbar<<<NB, CH, 0, stream>>>(srow, W, b, kbar);
    // 4) q projection via WMMA (16384 waves, 8 waves/block)
    {
        unsigned waves = (unsigned)(NROWS / 16) * 16;   // 1024 row tiles * 16 col tiles
        k_qproj_wmma<<<waves * 32 / 256, 256, 0, stream>>>(xn, W, b, q);
    }
    // 5) scores (wave per row)
    k_scores<<<(unsigned)(NROWS * 32) / 256, 256, 0, stream>>>(q, kbar, scores);
    // 6) top-k
    k_topk<<<NB, TT, 0, stream>>>(scores, topval, topidx);
    // 7) gated gather
    {
        unsigned total = NB * CH * NTP * VV;
        k_gather<<<(total + 255) / 256, 256, 0, stream>>>(x, topval, topidx, out);
    }
}